// QueryAndGroup_34574486733457
// MI455X (gfx1250) — compile-verified
//
#include <hip/hip_runtime.h>

// Problem constants (match setup_inputs in the reference)
#define B_   8
#define N_   8192
#define S_   2048
#define C_   64
#define NS_  32
#define CH_  (3 + C_)      // 67 output channels
#define R2_  0.01f         // radius^2

typedef __attribute__((ext_vector_type(2))) float v2f;
typedef __attribute__((ext_vector_type(8))) float v8f;

// ---------------------------------------------------------------------------
// Kernel A: ball query via V_WMMA_F32_16X16X4_F32.
// One wave (32 threads) handles 16 centroids; loops over 512 tiles of 16
// points. D[m,n] = -2*c_m.p_n + |p_n|^2 ; in-ball iff D < r^2 - |c_m|^2.
// Indices appended in point-index order via ballot + prefix popcount.
// Index list is stashed into the channel-66 slice of d_out (int bits).
// ---------------------------------------------------------------------------
__global__ __launch_bounds__(32)
void bq_wmma_kernel(const float* __restrict__ xyz,
                    const float* __restrict__ new_xyz,
                    int* __restrict__ idx_out)   // = (int*)d_out
{
    const int lane      = threadIdx.x;          // 0..31
    const int wg        = blockIdx.x;           // 0 .. B*(S/16)-1
    const int b         = wg / (S_ / 16);
    const int s0        = (wg % (S_ / 16)) * 16;
    const bool hi       = lane >= 16;
    const int  half_lane = lane & 15;

    __shared__ float csh[16][3];   // centroid coords
    __shared__ float thr[16];      // r^2 - |c|^2 per row
    __shared__ int   cnt[16];      // hits per row
    __shared__ int   first[16];    // first found index (0 if none)

    if (lane < 16) {
        const float* c = &new_xyz[((size_t)b * S_ + (s0 + lane)) * 3];
        float cx = c[0], cy = c[1], cz = c[2];
        csh[lane][0] = cx; csh[lane][1] = cy; csh[lane][2] = cz;
        thr[lane]   = R2_ - (cx * cx + cy * cy + cz * cz);
        cnt[lane]   = 0;
        first[lane] = 0;
    }
    __syncthreads();

    // A fragment (16x4 fp32): lanes 0-15 hold M=lane, K0/K1; lanes 16-31 K2/K3.
    v2f afrag;
    if (!hi) { afrag.x = -2.0f * csh[half_lane][0]; afrag.y = -2.0f * csh[half_lane][1]; }
    else     { afrag.x = -2.0f * csh[half_lane][2]; afrag.y = 1.0f; }

    // Per-lane thresholds for its 8 accumulator rows (m = r + 8*hi).
    float th[8];
    #pragma unroll
    for (int r = 0; r < 8; ++r) th[r] = thr[r + (hi ? 8 : 0)];

    const unsigned pre = (1u << half_lane) - 1u;   // lower-lane mask within half
    int* myidx = idx_out + (((size_t)b * CH_ + (CH_ - 1)) * S_ + s0) * NS_;

    for (int t = 0; t < N_ / 16; ++t) {
        const int n = t * 16 + half_lane;
        const float* p = &xyz[((size_t)b * N_ + n) * 3];
        float px = p[0], py = p[1], pz = p[2];

        // B fragment (4x16 fp32): lane = column N; low half K0/K1, high half K2/K3.
        v2f bfrag;
        if (!hi) { bfrag.x = px; bfrag.y = py; }
        else     { bfrag.x = pz; bfrag.y = px * px + py * py + pz * pz; }

        v8f acc = {};
        acc = __builtin_amdgcn_wmma_f32_16x16x4_f32(
                  /*neg_a=*/false, afrag, /*neg_b=*/false, bfrag,
                  /*c_mod=*/(short)0, acc, /*reuse_a=*/false, /*reuse_b=*/false);

        // Selection: acc[r] is D[m = r+8*hi, n = t*16 + lane%16].
        #pragma unroll
        for (int r = 0; r < 8; ++r) {
            bool inb = acc[r] < th[r];
            unsigned full = __builtin_amdgcn_ballot_w32(inb);
            unsigned hm = hi ? (full >> 16) : (full & 0xffffu);
            int m = r + (hi ? 8 : 0);
            int c = cnt[m];                      // all lanes read (in-order LDS)
            if (inb) {
                int pos = c + __popc(hm & pre);
                if (pos < NS_) myidx[(size_t)m * NS_ + pos] = n;
                if (pos == 0)  first[m] = n;     // exactly one lane, first hit
            }
            if (half_lane == 0) cnt[m] = c + __popc(hm);  // lane 0 / lane 16
        }

        if ((t & 31) == 31) {   // uniform early exit once all rows are full
            bool done = cnt[half_lane] >= NS_;
            if (__builtin_amdgcn_ballot_w32(done) == 0xffffffffu) break;
        }
    }
    __syncthreads();

    // Pad empty slots with first-found index (0 if none) — reference semantics.
    if (lane < 16) {
        int c = cnt[lane]; if (c > NS_) c = NS_;
        int fv = first[lane];
        for (int k = c; k < NS_; ++k) myidx[(size_t)lane * NS_ + k] = fv;
    }
}

// ---------------------------------------------------------------------------
// Kernel B: gather + recenter. One wave per (b, s); lane = sample k.
// Reads its 32 indices from the ch-66 stash first, then writes all 67
// channels (coalesced 128B stores per wave).
// ---------------------------------------------------------------------------
__global__ __launch_bounds__(256)
void group_kernel(const float* __restrict__ xyz,
                  const float* __restrict__ new_xyz,
                  const float* __restrict__ feat,
                  float* __restrict__ out)
{
    const int wid  = blockIdx.x * (blockDim.x >> 5) + (threadIdx.x >> 5);
    const int lane = threadIdx.x & 31;
    const int b = wid >> 11;          // / S_
    const int s = wid & (S_ - 1);

    const int* idxbuf =
        (const int*)out + (((size_t)b * CH_ + (CH_ - 1)) * S_ + s) * NS_;
    const int id = idxbuf[lane];      // read stash before any channel write

    const float* c = &new_xyz[((size_t)b * S_ + s) * 3];
    const float cx = c[0], cy = c[1], cz = c[2];

    const float* p = &xyz[((size_t)b * N_ + id) * 3];
    const float vx = p[0] - cx, vy = p[1] - cy, vz = p[2] - cz;

    float* o = out + (((size_t)b * CH_) * S_ + s) * NS_ + lane;
    const size_t chs = (size_t)S_ * NS_;   // channel stride

    o[0]       = vx;
    o[chs]     = vy;
    o[2 * chs] = vz;

    const float* fb = feat + ((size_t)b * C_) * N_ + id;
    #pragma unroll 4
    for (int ch = 0; ch < C_; ++ch)
        o[(size_t)(3 + ch) * chs] = fb[(size_t)ch * N_];
}

// ---------------------------------------------------------------------------
extern "C" void kernel_launch(void* const* d_in, const int* in_sizes, int n_in,
                              void* d_out, int out_size, void* d_ws, size_t ws_size,
                              hipStream_t stream) {
    (void)in_sizes; (void)n_in; (void)out_size; (void)d_ws; (void)ws_size;
    const float* xyz     = (const float*)d_in[0];   // (B, N, 3)
    const float* new_xyz = (const float*)d_in[1];   // (B, S, 3)
    const float* feat    = (const float*)d_in[2];   // (B, C, N)
    float* out = (float*)d_out;                     // (B, 67, S, 32)

    bq_wmma_kernel<<<B_ * (S_ / 16), 32, 0, stream>>>(xyz, new_xyz, (int*)out);
    group_kernel<<<(B_ * S_) / 8, 256, 0, stream>>>(xyz, new_xyz, feat, out);
}